// MeshAutoencoder_11029476016241
// MI455X (gfx1250) — compile-verified
//
#include <hip/hip_runtime.h>
#include <math.h>

// Problem constants (from reference: B, NV, NF = 8, 50000, 100000)
constexpr int   kB   = 8;
constexpr int   kNV  = 50000;
constexpr int   kNF  = 100000;
constexpr int   kD   = 21;          // 9 coords + 3 angles + 1 area + 3 normal + 1 emnoangle + 3 emangle + 1 emfreq
constexpr float kEps = 1e-5f;
constexpr float kDeg2Rad = 0.017453292519943295f;

static_assert((kB * kNV) % 16 == 0, "tile layout requires rows % 16 == 0");
static_assert(kNV % 16 == 0, "tiles must not straddle batches");

typedef float v2f __attribute__((ext_vector_type(2)));
typedef float v8f __attribute__((ext_vector_type(8)));

// ---------------------------------------------------------------------------
// Pass 0: zero the accumulators (d_out = feature sums, d_ws = incidence counts)
// ---------------------------------------------------------------------------
__global__ void ma_init_zero(float* __restrict__ out, float* __restrict__ cnt) {
  const int stride = gridDim.x * blockDim.x;
  const int tid = blockIdx.x * blockDim.x + threadIdx.x;
  const int nOut = kB * kNV * kD;
  const int nCnt = kB * kNV;
  for (int i = tid; i < nOut; i += stride) out[i] = 0.0f;
  for (int i = tid; i < nCnt; i += stride) cnt[i] = 0.0f;
}

__device__ __forceinline__ void l2norm3(float x, float y, float z,
                                        float& ox, float& oy, float& oz) {
  // reference: v / clip(||v||, 1e-12)
  float d = fmaxf(sqrtf(x * x + y * y + z * z), 1e-12f);
  ox = x / d; oy = y / d; oz = z / d;
}

__device__ __forceinline__ float clampAcos(float v) {
  return acosf(fminf(fmaxf(v, -1.0f + kEps), 1.0f - kEps));
}

// ---------------------------------------------------------------------------
// Pass 1: one thread per face. Gather 3 vertices, build the 21-dim embedding,
// scatter-add it (plus a count of 1) onto each of the face's 3 vertices.
// Non-returning global_atomic_add_f32 -> STOREcnt path, no waits needed.
// ---------------------------------------------------------------------------
__global__ void ma_face_scatter(const float* __restrict__ vertices,
                                const float* __restrict__ in_em,
                                const int*   __restrict__ faces,
                                float* __restrict__ out,
                                float* __restrict__ cnt) {
  const int t = blockIdx.x * blockDim.x + threadIdx.x;
  if (t >= kB * kNF) return;
  const int b = t / kNF;

  const int i0 = faces[3 * (size_t)t + 0];
  const int i1 = faces[3 * (size_t)t + 1];
  const int i2 = faces[3 * (size_t)t + 2];

  const float* vb = vertices + (size_t)b * kNV * 3;
  const float v0x = vb[3 * (size_t)i0 + 0], v0y = vb[3 * (size_t)i0 + 1], v0z = vb[3 * (size_t)i0 + 2];
  const float v1x = vb[3 * (size_t)i1 + 0], v1y = vb[3 * (size_t)i1 + 1], v1z = vb[3 * (size_t)i1 + 2];
  const float v2x = vb[3 * (size_t)i2 + 0], v2y = vb[3 * (size_t)i2 + 1], v2z = vb[3 * (size_t)i2 + 2];

  // edges = face_coords - roll(face_coords, 1, axis=2)
  const float e0x = v0x - v2x, e0y = v0y - v2y, e0z = v0z - v2z;
  const float e1x = v1x - v0x, e1y = v1y - v0y, e1z = v1z - v0z;
  const float e2x = v2x - v1x, e2y = v2y - v1y, e2z = v2z - v1z;

  float n0x, n0y, n0z, n1x, n1y, n1z, n2x, n2y, n2z;
  l2norm3(e0x, e0y, e0z, n0x, n0y, n0z);
  l2norm3(e1x, e1y, e1z, n1x, n1y, n1z);
  l2norm3(e2x, e2y, e2z, n2x, n2y, n2z);

  // normdot = -(normv * roll(normv, 1, axis=3)).sum(axis=2)  [roll over COORDS]
  const float nd0 = -(n0x * n0z + n1x * n1z + n2x * n2z);
  const float nd1 = -(n0y * n0x + n1y * n1x + n2y * n2x);
  const float nd2 = -(n0z * n0y + n1z * n1y + n2z * n2y);
  const float a0 = clampAcos(nd0);
  const float a1 = clampAcos(nd1);
  const float a2 = clampAcos(nd2);

  // cross(edge0, edge1)
  const float cx = e0y * e1z - e0z * e1y;
  const float cy = e0z * e1x - e0x * e1z;
  const float cz = e0x * e1y - e0y * e1x;
  const float cn = sqrtf(cx * cx + cy * cy + cz * cz);
  const float area = cn * 0.5f;
  float nx, ny, nz;
  { const float d = fmaxf(cn, 1e-12f); nx = cx / d; ny = cy / d; nz = cz / d; }

  const float theta = in_em[3 * b + 0] * kDeg2Rad;
  const float phi   = in_em[3 * b + 1] * kDeg2Rad;
  const float freq  = in_em[3 * b + 2];
  const float sp = sinf(phi);
  const float ix = sp * cosf(theta), iy = sp * sinf(theta), iz = cosf(phi);

  // reference re-normalizes both before the dot
  float nnx, nny, nnz, inx, iny, inz;
  l2norm3(nx, ny, nz, nnx, nny, nnz);
  l2norm3(ix, iy, iz, inx, iny, inz);
  const float dot = -(nnx * inx + nny * iny + nnz * inz);
  const float emno = clampAcos(dot);

  const float emb[kD] = { v0x, v0y, v0z, v1x, v1y, v1z, v2x, v2y, v2z,
                          a0, a1, a2, area, nx, ny, nz, emno, ix, iy, iz, freq };

  const int vids[3] = { i0, i1, i2 };
  #pragma unroll
  for (int c = 0; c < 3; ++c) {
    const size_t base = ((size_t)b * kNV + (size_t)vids[c]) * kD;
    #pragma unroll
    for (int k = 0; k < kD; ++k) atomicAdd(out + base + k, emb[k]);
    atomicAdd(cnt + (size_t)b * kNV + (size_t)vids[c], 1.0f);
  }
}

// ---------------------------------------------------------------------------
// Pass 2: out[row][k] = sum[row][k] * (1 / max(count[row], EPS)).
// Columns 0..15 of each 16-row tile are done as D = diag(r) * S via
// V_WMMA_F32_16X16X4_F32 (4 chained K=4 slices). Only one nonzero product per
// output element, so the result is exactly RNE(r*s). Columns 16..20 via VALU.
// One wave per tile; grid exactly covers all tiles so EXEC is all ones.
// ---------------------------------------------------------------------------
__global__ void ma_finalize_scale(float* __restrict__ out, const float* __restrict__ cnt) {
  const int lane = threadIdx.x & 31;
  const int wave = threadIdx.x >> 5;
  const int tile = blockIdx.x * (blockDim.x >> 5) + wave;
  const int totalTiles = (kB * kNV) / 16;   // 25000
  if (tile >= totalTiles) return;           // wave-uniform; never taken with exact grid
  const int rowBase = tile * 16;

  const int  m  = lane & 15;                // my output row within tile (A layout: lane -> M)
  const int  n  = lane & 15;                // my column within tile (B/C/D layout: lane -> N)
  const bool hi = lane >= 16;

  // reciprocal of clipped incidence count for row m
  const float r = 1.0f / fmaxf(cnt[rowBase + m], kEps);

  v8f c = {};
  #pragma unroll
  for (int kb = 0; kb < 4; ++kb) {
    // B layout (mirrors documented A layout): lanes 0-15 hold K = 4*kb+{0,1},
    // lanes 16-31 hold K = 4*kb+{2,3}; lane%16 selects column N.
    const int kr0 = 4 * kb + (hi ? 2 : 0);
    const int kr1 = kr0 + 1;
    v2f bm, am;
    bm.x = out[(size_t)(rowBase + kr0) * kD + n];
    bm.y = out[(size_t)(rowBase + kr1) * kD + n];
    // A = K-slice of diag(r): A[m][k] = (m == k_global) ? r : 0
    am.x = (m == kr0) ? r : 0.0f;
    am.y = (m == kr1) ? r : 0.0f;
    c = __builtin_amdgcn_wmma_f32_16x16x4_f32(
        /*neg_a=*/false, am, /*neg_b=*/false, bm,
        /*c_mod=*/(short)0, c, /*reuse_a=*/false, /*reuse_b=*/false);
  }

  // C/D layout: VGPR j, lanes 0-15 -> row j, lanes 16-31 -> row j+8, col = lane%16
  #pragma unroll
  for (int j = 0; j < 8; ++j) {
    const int row = rowBase + j + (hi ? 8 : 0);
    out[(size_t)row * kD + n] = c[j];
  }

  // Remainder columns 16..20 (16 rows x 5 cols) with plain VALU
  for (int idx = lane; idx < 16 * 5; idx += 32) {
    const int rr = idx / 5;
    const int cc = 16 + idx % 5;
    const float rv = 1.0f / fmaxf(cnt[rowBase + rr], kEps);
    const size_t o = (size_t)(rowBase + rr) * kD + cc;
    out[o] = out[o] * rv;
  }
}

// ---------------------------------------------------------------------------
extern "C" void kernel_launch(void* const* d_in, const int* in_sizes, int n_in,
                              void* d_out, int out_size, void* d_ws, size_t ws_size,
                              hipStream_t stream) {
  const float* vertices = (const float*)d_in[0];   // (B, NV, 3) f32
  const float* in_em    = (const float*)d_in[1];   // (B, 3)     f32
  const int*   faces    = (const int*)  d_in[2];   // (B, NF, 3) i32
  float* out = (float*)d_out;                      // (B, NV, 21) f32 sums -> result
  float* cnt = (float*)d_ws;                       // (B, NV) f32 incidence counts

  hipLaunchKernelGGL(ma_init_zero, dim3(2048), dim3(256), 0, stream, out, cnt);

  const int nFaces = kB * kNF;                     // 800000
  hipLaunchKernelGGL(ma_face_scatter, dim3((nFaces + 255) / 256), dim3(256), 0, stream,
                     vertices, in_em, faces, out, cnt);

  const int tiles = (kB * kNV) / 16;               // 25000
  const int wavesPerBlock = 8;                     // 256 threads = 8 wave32
  hipLaunchKernelGGL(ma_finalize_scale,
                     dim3((tiles + wavesPerBlock - 1) / wavesPerBlock), dim3(256), 0, stream,
                     out, cnt);
}